// SimpleResonantLayer_75892072121040
// MI455X (gfx1250) — compile-verified
//
#include <hip/hip_runtime.h>
#include <math.h>

typedef __attribute__((ext_vector_type(2))) float v2f;
typedef __attribute__((ext_vector_type(4))) float v4f;
typedef __attribute__((ext_vector_type(8))) float v8f;
typedef __attribute__((ext_vector_type(4))) int   v4i;

#define LUT_N     4096
#define LUT_MASK  4095
#define TWO_PI_F  6.2831853071795864769f
#define LUT_SCALE 651.8986469044033f   /* 4096 / (2*pi) */

#define BATCH 32
#define DMOD  1024
#define NRES  2048

// ---------------------------------------------------------------------------
// CDNA5 async global->LDS helpers (ASYNCcnt path), with safe fallbacks.
// Builtin signature (from hipcc diagnostic): param0 = AS1 int4*, so cast
// through <4 x i32> pointers in the explicit address spaces.
// ---------------------------------------------------------------------------
typedef __attribute__((address_space(1))) v4i* glob_v4i_ptr;
typedef __attribute__((address_space(3))) v4i* lds_v4i_ptr;

__device__ __forceinline__ void async_copy16(const float* g, float* l) {
#if __has_builtin(__builtin_amdgcn_global_load_async_to_lds_b128)
    __builtin_amdgcn_global_load_async_to_lds_b128(
        (glob_v4i_ptr)g, (lds_v4i_ptr)l, 0, 0);
#else
    *(v4f*)l = *(const v4f*)g;
#endif
}

__device__ __forceinline__ void wait_async0() {
#if __has_builtin(__builtin_amdgcn_s_wait_asynccnt)
    __builtin_amdgcn_s_wait_asynccnt(0);
#elif __has_builtin(__builtin_amdgcn_global_load_async_to_lds_b128)
    asm volatile("s_wait_asynccnt 0" ::: "memory");
#endif
}

// ---------------------------------------------------------------------------
// Kernel 1: build interleaved {sin, cos} LUT in workspace
// ---------------------------------------------------------------------------
__global__ __launch_bounds__(256) void lut_init_kernel(float2* __restrict__ lut) {
    int i = blockIdx.x * blockDim.x + threadIdx.x;
    if (i < LUT_N) {
        float ang = (float)i * (TWO_PI_F / (float)LUT_N);
        lut[i] = make_float2(sinf(ang), cosf(ang));
    }
}

// ---------------------------------------------------------------------------
// Kernel 2: input projection, fp32 WMMA. Writes xc TRANSPOSED: xcT[j, m],
// shape (1024, 32), so the theta kernel can async-stage contiguous d-slabs
// and read conflict-free columns.
//   xc[m, j] = sum_k concat(x_r, x_i)[m, k] * Wp[j, k] + bias[j]
// ---------------------------------------------------------------------------
__global__ __launch_bounds__(128) void proj_in_wmma_kernel(
        const float* __restrict__ xr, const float* __restrict__ xi,
        const float* __restrict__ Wp, const float* __restrict__ bias,
        float* __restrict__ xcT) {
    const int K = 2 * DMOD;                       // 2048
    int wid   = blockIdx.x * (blockDim.x >> 5) + (threadIdx.x >> 5);
    int lane  = threadIdx.x & 31;
    int m0    = (wid & 1) * 16;                   // batch tile
    int n0    = (wid >> 1) * 16;                  // output-feature tile
    int mh    = lane & 15;                        // A row (M) / B col (N)
    int khalf = lane >> 4;                        // 0: K=0,1  1: K=2,3

    const float* arow_r = xr + (size_t)(m0 + mh) * DMOD;
    const float* arow_i = xi + (size_t)(m0 + mh) * DMOD;
    const float* brow   = Wp + (size_t)(n0 + mh) * K;

    v8f acc0 = {}; v8f acc1 = {};
    // K in [0, 1024): A from x_real
    for (int k = 0; k < DMOD; k += 8) {
        int kb0 = k + khalf * 2, kb1 = kb0 + 4;
        v2f a0; a0.x = arow_r[kb0]; a0.y = arow_r[kb0 + 1];
        v2f b0; b0.x = brow[kb0];   b0.y = brow[kb0 + 1];
        v2f a1; a1.x = arow_r[kb1]; a1.y = arow_r[kb1 + 1];
        v2f b1; b1.x = brow[kb1];   b1.y = brow[kb1 + 1];
        acc0 = __builtin_amdgcn_wmma_f32_16x16x4_f32(false, a0, false, b0,
                                                     (short)0, acc0, false, false);
        acc1 = __builtin_amdgcn_wmma_f32_16x16x4_f32(false, a1, false, b1,
                                                     (short)0, acc1, false, false);
    }
    // K in [1024, 2048): A from x_imag
    for (int k = DMOD; k < K; k += 8) {
        int kb0 = k + khalf * 2, kb1 = kb0 + 4;
        v2f a0; a0.x = arow_i[kb0 - DMOD]; a0.y = arow_i[kb0 - DMOD + 1];
        v2f b0; b0.x = brow[kb0];          b0.y = brow[kb0 + 1];
        v2f a1; a1.x = arow_i[kb1 - DMOD]; a1.y = arow_i[kb1 - DMOD + 1];
        v2f b1; b1.x = brow[kb1];          b1.y = brow[kb1 + 1];
        acc0 = __builtin_amdgcn_wmma_f32_16x16x4_f32(false, a0, false, b0,
                                                     (short)0, acc0, false, false);
        acc1 = __builtin_amdgcn_wmma_f32_16x16x4_f32(false, a1, false, b1,
                                                     (short)0, acc1, false, false);
    }
    // C/D layout: VGPR r -> M = r + 8*(lane>=16), N = lane & 15
    int mrow = (lane >> 4) * 8;
    int ncol = lane & 15;
    float bv = bias[n0 + ncol];
    #pragma unroll
    for (int r = 0; r < 8; ++r) {
        float v = acc0[r] + acc1[r] + bv;
        xcT[(size_t)(n0 + ncol) * BATCH + (m0 + mrow + r)] = v;   // transposed
    }
}

// ---------------------------------------------------------------------------
// Kernel 3: heavy resonant layer, double-buffered async LDS staging.
//   fidx[b,n,d] = xcT[d,b] * (S/(1+|W[n,d]|)) + B[n,d]*S + t[b]*S
//   LUT lerp sin/cos, reduce over d.
//   Block: TN=8 n-rows, 128 threads; thread = (nl, bg) handles b in {bg, bg+16}.
// ---------------------------------------------------------------------------
#define TN 8
#define DC 256
#define NCHUNK (DMOD / DC)

__global__ __launch_bounds__(128) void theta_sum_kernel(
        const float*  __restrict__ xcT, const float* __restrict__ W,
        const float*  __restrict__ Bm,  const float* __restrict__ t,
        const float2* __restrict__ lut_g,
        float* __restrict__ cos_sum, float* __restrict__ sin_sum) {
    __shared__ float2 lut_s[LUT_N];            // 32 KB interleaved {sin, cos}
    __shared__ float  xc_s[2][DC * BATCH];     // 2 x 32 KB, [d][b] slabs
    __shared__ float  w_s[2][TN * DC];         // 2 x 8 KB, becomes S/(1+|W|)
    __shared__ float  b_s[2][TN * DC];         // 2 x 8 KB, becomes B*S

    const int tid = threadIdx.x;
    const int n0  = blockIdx.x * TN;
    const int nl  = tid >> 4;                  // 0..7 (2 values per wave)
    const int bg  = tid & 15;                  // lanes -> consecutive banks

    for (int i = tid; i < LUT_N; i += 128) lut_s[i] = lut_g[i];

    const float tb0 = t[bg] * LUT_SCALE;
    const float tb1 = t[bg + 16] * LUT_SCALE;
    float s0 = 0.f, c0 = 0.f, s1 = 0.f, c1 = 0.f;

    auto issue_chunk = [&](int c, int buf) {
        // xcT d-slab is fully contiguous: DC*BATCH floats
        const float* xg = xcT + (size_t)c * DC * BATCH;
        for (int u = tid; u < DC * BATCH / 4; u += 128)
            async_copy16(xg + u * 4, &xc_s[buf][u * 4]);
        // W/B: TN rows of DC contiguous floats (row stride DMOD)
        for (int u = tid; u < TN * DC / 4; u += 128) {
            int row  = u >> 6;                 // DC/4 == 64 b128 units per row
            int col  = (u & 63) << 2;
            size_t g = (size_t)(n0 + row) * DMOD + (size_t)c * DC + col;
            async_copy16(W  + g, &w_s[buf][u * 4]);   // row*DC + col == u*4
            async_copy16(Bm + g, &b_s[buf][u * 4]);
        }
    };
    auto transform_chunk = [&](int buf) {
        // each thread transforms exactly the region it async-loaded
        for (int u = tid; u < TN * DC / 4; u += 128) {
            int idx = u * 4;
            v4f wv = *(v4f*)&w_s[buf][idx];
            v4f bv = *(v4f*)&b_s[buf][idx];
            wv.x = LUT_SCALE / (1.0f + fabsf(wv.x));
            wv.y = LUT_SCALE / (1.0f + fabsf(wv.y));
            wv.z = LUT_SCALE / (1.0f + fabsf(wv.z));
            wv.w = LUT_SCALE / (1.0f + fabsf(wv.w));
            bv.x *= LUT_SCALE; bv.y *= LUT_SCALE;
            bv.z *= LUT_SCALE; bv.w *= LUT_SCALE;
            *(v4f*)&w_s[buf][idx] = wv;
            *(v4f*)&b_s[buf][idx] = bv;
        }
    };

    issue_chunk(0, 0);
    wait_async0();
    transform_chunk(0);
    __syncthreads();

    for (int c = 0; c < NCHUNK; ++c) {
        const int cur = c & 1, nxt = cur ^ 1;
        if (c + 1 < NCHUNK) issue_chunk(c + 1, nxt);   // overlap with compute

        const float* xcp = xc_s[cur];
        const float* wp  = &w_s[cur][nl * DC];
        const float* bp  = &b_s[cur][nl * DC];
        #pragma unroll 4
        for (int d = 0; d < DC; ++d) {
            float rws = wp[d];                  // wave broadcast
            float bbs = bp[d];
            float x0  = xcp[d * BATCH + bg];    // conflict-free row read
            float x1  = xcp[d * BATCH + bg + 16];
            {
                float f  = fmaf(x0, rws, bbs) + tb0;
                float fl = floorf(f);
                float w  = f - fl;
                int i0 = ((int)fl) & LUT_MASK;  // pow2 mod handles negatives
                int i1 = (i0 + 1) & LUT_MASK;
                float2 e0 = lut_s[i0], e1 = lut_s[i1];
                s0 = fmaf(e1.x - e0.x, w, s0 + e0.x);
                c0 = fmaf(e1.y - e0.y, w, c0 + e0.y);
            }
            {
                float f  = fmaf(x1, rws, bbs) + tb1;
                float fl = floorf(f);
                float w  = f - fl;
                int i0 = ((int)fl) & LUT_MASK;
                int i1 = (i0 + 1) & LUT_MASK;
                float2 e0 = lut_s[i0], e1 = lut_s[i1];
                s1 = fmaf(e1.x - e0.x, w, s1 + e0.x);
                c1 = fmaf(e1.y - e0.y, w, c1 + e0.y);
            }
        }
        if (c + 1 < NCHUNK) {
            wait_async0();                      // next chunk landed in LDS
            transform_chunk(nxt);               // in-place, own region only
        }
        __syncthreads();
    }

    const int n = n0 + nl;
    cos_sum[(size_t)bg * NRES + n]        = c0;
    sin_sum[(size_t)bg * NRES + n]        = s0;
    cos_sum[(size_t)(bg + 16) * NRES + n] = c1;
    sin_sum[(size_t)(bg + 16) * NRES + n] = s1;
}

// ---------------------------------------------------------------------------
// Kernel 4: output projection + SiLU, fp32 WMMA.
//   out[m, j] = silu( sum_k A[m, k] * Wo[j, k] ),  M=32, K=2048, Nout=1024
// ---------------------------------------------------------------------------
__global__ __launch_bounds__(128) void proj_out_wmma_kernel(
        const float* __restrict__ A, const float* __restrict__ Wo,
        float* __restrict__ out) {
    const int K = NRES;                           // 2048
    int wid   = blockIdx.x * (blockDim.x >> 5) + (threadIdx.x >> 5);
    int lane  = threadIdx.x & 31;
    int m0    = (wid & 1) * 16;
    int n0    = (wid >> 1) * 16;
    int mh    = lane & 15;
    int khalf = lane >> 4;

    const float* arow = A  + (size_t)(m0 + mh) * K;
    const float* brow = Wo + (size_t)(n0 + mh) * K;

    v8f acc0 = {}; v8f acc1 = {};
    for (int k = 0; k < K; k += 8) {
        int kb0 = k + khalf * 2, kb1 = kb0 + 4;
        v2f a0; a0.x = arow[kb0]; a0.y = arow[kb0 + 1];
        v2f b0; b0.x = brow[kb0]; b0.y = brow[kb0 + 1];
        v2f a1; a1.x = arow[kb1]; a1.y = arow[kb1 + 1];
        v2f b1; b1.x = brow[kb1]; b1.y = brow[kb1 + 1];
        acc0 = __builtin_amdgcn_wmma_f32_16x16x4_f32(false, a0, false, b0,
                                                     (short)0, acc0, false, false);
        acc1 = __builtin_amdgcn_wmma_f32_16x16x4_f32(false, a1, false, b1,
                                                     (short)0, acc1, false, false);
    }
    int mrow = (lane >> 4) * 8;
    int ncol = lane & 15;
    #pragma unroll
    for (int r = 0; r < 8; ++r) {
        float v = acc0[r] + acc1[r];
        float sig = 1.0f / (1.0f + __expf(-v));
        out[(size_t)(m0 + mrow + r) * DMOD + (n0 + ncol)] = v * sig;
    }
}

// ---------------------------------------------------------------------------
// Launch
// ---------------------------------------------------------------------------
extern "C" void kernel_launch(void* const* d_in, const int* in_sizes, int n_in,
                              void* d_out, int out_size, void* d_ws, size_t ws_size,
                              hipStream_t stream) {
    const float* x_real = (const float*)d_in[0];   // (32, 1024)
    const float* x_imag = (const float*)d_in[1];   // (32, 1024)
    const float* t      = (const float*)d_in[2];   // (32,)
    const float* ipw    = (const float*)d_in[3];   // (1024, 2048)
    const float* ipb    = (const float*)d_in[4];   // (1024,)
    const float* W      = (const float*)d_in[5];   // (2048, 1024)
    const float* Bm     = (const float*)d_in[6];   // (2048, 1024)
    const float* orw    = (const float*)d_in[7];   // (1024, 2048)
    const float* oiw    = (const float*)d_in[8];   // (1024, 2048)

    float*  ws  = (float*)d_ws;
    float2* lut = (float2*)ws;                       // 4096 float2  (32 KB)
    float*  xcT = ws + 2 * LUT_N;                    // (1024, 32)   (128 KB)
    float*  cs  = xcT + DMOD * BATCH;                // (32, 2048)   (256 KB)
    float*  ss  = cs + BATCH * NRES;                 // (32, 2048)   (256 KB)

    float* out_real = (float*)d_out;                 // (32, 1024)
    float* out_imag = out_real + BATCH * DMOD;       // (32, 1024)

    lut_init_kernel<<<LUT_N / 256, 256, 0, stream>>>(lut);
    proj_in_wmma_kernel<<<32, 128, 0, stream>>>(x_real, x_imag, ipw, ipb, xcT);
    theta_sum_kernel<<<NRES / TN, 128, 0, stream>>>(xcT, W, Bm, t, lut, cs, ss);
    proj_out_wmma_kernel<<<32, 128, 0, stream>>>(cs, orw, out_real);
    proj_out_wmma_kernel<<<32, 128, 0, stream>>>(ss, oiw, out_imag);
}